// DistributedAFNO2D_824633720960
// MI455X (gfx1250) — compile-verified
//
#include <hip/hip_runtime.h>

typedef __attribute__((ext_vector_type(2))) float v2f;
typedef __attribute__((ext_vector_type(8))) float v8f;
typedef __attribute__((ext_vector_type(4))) unsigned int v4u;
typedef __attribute__((ext_vector_type(8))) int v8i;
typedef __attribute__((ext_vector_type(4))) int v4i;

#define DEVI __device__ __forceinline__

namespace cfg {
constexpr int H = 180, W = 360, C = 768, NB = 8, BS = 96;
constexpr int WF = 91;                  // kept rFFT modes along W (rest are zeroed by ref)
constexpr int SH = WF * H;              // 16380 spatial modes per channel
constexpr int CH = C * H;               // 138240 rows for W-direction transforms
constexpr int ROWH = C * WF;            // 69888 rows for H-direction transforms
constexpr int PLANE = C * SH;           // 12579840 floats per re/im plane
// basis region (floats), N-dims padded to WMMA tiles and zero-filled
constexpr int SB1 = W * 96;             // 34560  (cos/-sin for rfft W, N padded 91->96)
constexpr int SB2 = H * 192;            // 34560  (fwd H DFT, N padded 180->192, ortho scale folded)
constexpr int SB4 = 92 * 368;           // 33856  (irfft W, K padded 91->92, N padded 360->368)
constexpr int OB1C = 0;
constexpr int OB1S = OB1C + SB1;
constexpr int OB2C = OB1S + SB1;
constexpr int OB2S = OB2C + SB2;
constexpr int OB3C = OB2S + SB2;
constexpr int OB3S = OB3C + SB2;
constexpr int OB4C = OB3S + SB2;
constexpr int OB4S = OB4C + SB4;
constexpr int OP0R = OB4S + SB4;
constexpr int OP0I = OP0R + PLANE;
constexpr int OP1R = OP0I + PLANE;
constexpr int OP1I = OP1R + PLANE;
constexpr long long TOTWS = (long long)OP1I + PLANE;  // ~50.6M floats ~= 193 MiB
}

// ---------------- WMMA helper: D = A(16x4,f32) * B(4x16,f32) + C --------------
DEVI v8f wmma4(v2f a, v2f b, v8f c) {
  return __builtin_amdgcn_wmma_f32_16x16x4_f32(false, a, false, b, (short)0, c,
                                               false, false);
}

DEVI float sshrink(float v) {
  const float L = 0.01f;
  return v > L ? v - L : (v < -L ? v + L : 0.0f);
}

// ---------------- TDM: 1-D tensor copy global -> LDS (6-arg clang-23 form) ----
DEVI void tdm_copy_1d(const void* gsrc, unsigned lds_byte_addr, unsigned nElems) {
  unsigned long long ga = (unsigned long long)(size_t)gsrc;
  v4u g0;
  g0[0] = 1u;                                          // count=1, user D#
  g0[1] = lds_byte_addr;                               // lds_addr  [63:32]
  g0[2] = (unsigned)(ga & 0xFFFFFFFFu);                // global_addr[31:0]
  g0[3] = (unsigned)((ga >> 32) & 0x1FFFFFFu) | (2u << 30);  // addr[56:32]|type=2
  v8i g1;
  g1[0] = (int)(2u << 16);        // workgroup_mask=0, data_size=2 (4 bytes)
  g1[1] = (int)(nElems << 16);    // tensor_dim0[15:0]  @bits[63:48]
  g1[2] = (int)(1u << 16);        // tensor_dim0 hi=0 | tensor_dim1=1
  g1[3] = (int)(nElems << 16);    // tile_dim0          @bits[127:112]
  g1[4] = 0;                      // tile_dim1=0 (unused), tile_dim2=0
  g1[5] = (int)nElems;            // tensor_dim0_stride lo
  g1[6] = 0;
  g1[7] = 0;
  v4i z4 = {0, 0, 0, 0};
  v8i z8 = {0, 0, 0, 0, 0, 0, 0, 0};
  __builtin_amdgcn_tensor_load_to_lds(g0, g1, z4, z4, z8, 0);
}

// ---------------- basis generation -------------------------------------------
__global__ __launch_bounds__(256) void init_basis_kernel(
    float* B1c, float* B1s, float* B2c, float* B2s, float* B3c, float* B3s,
    float* B4c, float* B4s) {
  int i = blockIdx.x * blockDim.x + threadIdx.x;
  if (i >= 34560) return;
  const double TP = 6.283185307179586476925286766559;
  const double sf = 1.0 / sqrt(64800.0);  // 1/sqrt(H*W), one 'ortho' factor
  {  // B1: rfft along W.  Re uses cos, Im uses -sin.   [w][n], n<91 valid
    int w = i / 96, n = i % 96;
    double th = TP * (double)((w * n) % 360) / 360.0;
    bool v = n < cfg::WF;
    B1c[i] = v ? (float)cos(th) : 0.0f;
    B1s[i] = v ? (float)(-sin(th)) : 0.0f;
  }
  {  // B2: forward DFT along H (ortho scale folded).  B3: inverse (unscaled)
    int h = i / 192, n = i % 192;
    bool v = n < cfg::H;
    double th = TP * (double)((h * n) % 180) / 180.0;
    double cv = cos(th), sv = sin(th);
    B2c[i] = v ? (float)(sf * cv) : 0.0f;
    B2s[i] = v ? (float)(sf * sv) : 0.0f;
    B3c[i] = v ? (float)cv : 0.0f;
    B3s[i] = v ? (float)sv : 0.0f;
  }
  if (i < cfg::SB4) {  // B4: irfft along W with hermitian weights + ortho scale
    int k = i / 368, w = i % 368;
    bool v = (k < cfg::WF) && (w < cfg::W);
    double al = (k == 0) ? 1.0 : 2.0;
    double th = TP * (double)((k * w) % 360) / 360.0;
    B4c[i] = v ? (float)(al * sf * cos(th)) : 0.0f;
    B4s[i] = v ? (float)(-al * sf * sin(th)) : 0.0f;
  }
}

// ---------------- stage 1: rFFT along W as WMMA GEMM --------------------------
// in : x rows (c*180+h) x 360 real
// out: planes (c, wf<91, h):  P0r/P0i[(c*91+wf)*180 + h]
__global__ __launch_bounds__(256) void fft_w_kernel(
    const float* __restrict__ x, const float* __restrict__ bc,
    const float* __restrict__ bs, float* __restrict__ outR,
    float* __restrict__ outI) {
  int wave = (int)((blockIdx.x * blockDim.x + threadIdx.x) >> 5);
  int lane = threadIdx.x & 31;
  int half = lane >> 4, l15 = lane & 15;
  int m0 = wave * 16;
  const float* arow = x + (size_t)(m0 + l15) * cfg::W;
  v8f zero = {0.f, 0.f, 0.f, 0.f, 0.f, 0.f, 0.f, 0.f};
  v8f accR[6], accI[6];
#pragma unroll
  for (int n = 0; n < 6; ++n) { accR[n] = zero; accI[n] = zero; }
  for (int k0 = 0; k0 < cfg::W; k0 += 4) {
    int ka = k0 + 2 * half;
    v2f a;
    a.x = arow[ka];
    a.y = arow[ka + 1];
#pragma unroll
    for (int n = 0; n < 6; ++n) {
      int nc = n * 16 + l15;
      const float* pc = bc + (size_t)ka * 96 + nc;
      const float* ps = bs + (size_t)ka * 96 + nc;
      v2f c; c.x = pc[0]; c.y = pc[96];
      v2f s; s.x = ps[0]; s.y = ps[96];
      accR[n] = wmma4(a, c, accR[n]);
      accI[n] = wmma4(a, s, accI[n]);
    }
  }
#pragma unroll
  for (int n = 0; n < 6; ++n) {
    int wf = n * 16 + l15;
    if (wf < cfg::WF) {
#pragma unroll
      for (int j = 0; j < 8; ++j) {
        int m = m0 + j + 8 * half;
        int c = m / cfg::H, h = m - c * cfg::H;
        size_t o = ((size_t)c * cfg::WF + wf) * cfg::H + h;
        outR[o] = accR[n][j];
        outI[o] = accI[n][j];
      }
    }
  }
}

// ---------------- stages 2 & 4: complex DFT along H ---------------------------
// MODE 0 fwd : Re = Ar*C + Ai*S ; Im = Ai*C - Ar*S      (C,S carry ortho scale)
// MODE 1 inv : Re = Ar*C - Ai*S ; Im = Ar*S + Ai*C
template <int MODE>
__global__ __launch_bounds__(256) void dft_h_kernel(
    const float* __restrict__ Ar, const float* __restrict__ Ai,
    const float* __restrict__ bc, const float* __restrict__ bs,
    float* __restrict__ Or, float* __restrict__ Oi) {
  int gw = (int)((blockIdx.x * blockDim.x + threadIdx.x) >> 5);
  int mt = gw >> 1, nh = gw & 1;  // each wave: one M tile, half of the N tiles
  int lane = threadIdx.x & 31;
  int half = lane >> 4, l15 = lane & 15;
  int m0 = mt * 16;
  const float* rr = Ar + (size_t)(m0 + l15) * cfg::H;
  const float* ri = Ai + (size_t)(m0 + l15) * cfg::H;
  v8f zero = {0.f, 0.f, 0.f, 0.f, 0.f, 0.f, 0.f, 0.f};
  v8f accR[6], accI[6];
#pragma unroll
  for (int n = 0; n < 6; ++n) { accR[n] = zero; accI[n] = zero; }
  for (int k0 = 0; k0 < cfg::H; k0 += 4) {
    int ka = k0 + 2 * half;
    v2f ar; ar.x = rr[ka]; ar.y = rr[ka + 1];
    v2f ai; ai.x = ri[ka]; ai.y = ri[ka + 1];
#pragma unroll
    for (int n = 0; n < 6; ++n) {
      int nc = (nh * 6 + n) * 16 + l15;
      const float* pc = bc + (size_t)ka * 192 + nc;
      const float* ps = bs + (size_t)ka * 192 + nc;
      v2f c; c.x = pc[0]; c.y = pc[192];
      v2f s; s.x = ps[0]; s.y = ps[192];
      v2f sn = -s;
      if (MODE == 0) {
        accR[n] = wmma4(ar, c, accR[n]);
        accR[n] = wmma4(ai, s, accR[n]);
        accI[n] = wmma4(ai, c, accI[n]);
        accI[n] = wmma4(ar, sn, accI[n]);
      } else {
        accR[n] = wmma4(ar, c, accR[n]);
        accR[n] = wmma4(ai, sn, accR[n]);
        accI[n] = wmma4(ar, s, accI[n]);
        accI[n] = wmma4(ai, c, accI[n]);
      }
    }
  }
#pragma unroll
  for (int n = 0; n < 6; ++n) {
    int np = (nh * 6 + n) * 16 + l15;
    if (np < cfg::H) {
#pragma unroll
      for (int j = 0; j < 8; ++j) {
        size_t o = (size_t)(m0 + j + 8 * half) * cfg::H + np;
        Or[o] = accR[n][j];
        Oi[o] = accI[n][j];
      }
    }
  }
}

// ---------------- stage 3: per-block complex MLP (WMMA, TDM-staged biases) ----
// planes: Y/Z[(kb*96+ch)*SH + s],  s = wf*180 + h'  (only wf<91 exists anywhere)
__global__ __launch_bounds__(128) void mlp_kernel(
    const float* __restrict__ Yr, const float* __restrict__ Yi,
    const float* __restrict__ w1, const float* __restrict__ b1,
    const float* __restrict__ w2, const float* __restrict__ b2,
    float* __restrict__ Zr, float* __restrict__ Zi) {
  __shared__ float sB1[192];
  __shared__ float sB2[192];
  __shared__ float sO1[4][16][96][2];  // per-wave D->A fragment bounce, 48 KB

  int kb = blockIdx.x & 7;
  int tg = blockIdx.x >> 3;  // 0..127 tile groups
  int wid = threadIdx.x >> 5;
  int lane = threadIdx.x & 31;
  int half = lane >> 4, l15 = lane & 15;
  int cb = kb * cfg::BS;
  const float* w1b = w1 + (size_t)kb * cfg::BS * cfg::BS * 2;
  const float* w2b = w2 + (size_t)kb * cfg::BS * cfg::BS * 2;

  // TDM-stage the per-block biases into LDS; wave 0 drives the DMA.
  if (wid == 0) {
    tdm_copy_1d(b1 + (size_t)kb * 192, (unsigned)(size_t)&sB1[0], 192);
    tdm_copy_1d(b2 + (size_t)kb * 192, (unsigned)(size_t)&sB2[0], 192);
    __builtin_amdgcn_s_wait_tensorcnt(0);
  }
  __builtin_prefetch(w1b, 0, 1);
  __builtin_prefetch(w2b, 0, 1);
  __syncthreads();

  v8f zero = {0.f, 0.f, 0.f, 0.f, 0.f, 0.f, 0.f, 0.f};
  for (int it = 0; it < 2; ++it) {  // 1024 M tiles = 128 groups * 4 waves * 2
    int tile = tg * 8 + wid * 2 + it;
    int s0 = tile * 16;
    int sm = s0 + l15;
    bool mv = sm < cfg::SH;

    // ---- layer 1: o1 = relu(Y x W1 + b1) ----
    v8f accR[6], accI[6];
#pragma unroll
    for (int n = 0; n < 6; ++n) { accR[n] = zero; accI[n] = zero; }
    for (int k0 = 0; k0 < cfg::BS; k0 += 4) {
      int ka = k0 + 2 * half;
      v2f ar, ai;
      ar.x = mv ? Yr[(size_t)(cb + ka) * cfg::SH + sm] : 0.f;
      ar.y = mv ? Yr[(size_t)(cb + ka + 1) * cfg::SH + sm] : 0.f;
      ai.x = mv ? Yi[(size_t)(cb + ka) * cfg::SH + sm] : 0.f;
      ai.y = mv ? Yi[(size_t)(cb + ka + 1) * cfg::SH + sm] : 0.f;
#pragma unroll
      for (int n = 0; n < 6; ++n) {
        int oc = n * 16 + l15;
        v2f wr, wi;
        wr.x = w1b[((size_t)ka * 96 + oc) * 2 + 0];
        wr.y = w1b[((size_t)(ka + 1) * 96 + oc) * 2 + 0];
        wi.x = w1b[((size_t)ka * 96 + oc) * 2 + 1];
        wi.y = w1b[((size_t)(ka + 1) * 96 + oc) * 2 + 1];
        v2f wn = -wi;
        accR[n] = wmma4(ar, wr, accR[n]);
        accR[n] = wmma4(ai, wn, accR[n]);
        accI[n] = wmma4(ai, wr, accI[n]);
        accI[n] = wmma4(ar, wi, accI[n]);
      }
    }
#pragma unroll
    for (int n = 0; n < 6; ++n) {
      int oc = n * 16 + l15;
      float br = sB1[oc * 2 + 0], bi = sB1[oc * 2 + 1];
#pragma unroll
      for (int j = 0; j < 8; ++j) {
        float r = accR[n][j] + br;
        float m = accI[n][j] + bi;
        sO1[wid][j + 8 * half][oc][0] = r > 0.f ? r : 0.f;
        sO1[wid][j + 8 * half][oc][1] = m > 0.f ? m : 0.f;
      }
    }
    __syncthreads();

    // ---- layer 2: z = softshrink(o1 x W2 + b2) ----
    v8f dR[6], dI[6];
#pragma unroll
    for (int n = 0; n < 6; ++n) { dR[n] = zero; dI[n] = zero; }
    for (int k0 = 0; k0 < cfg::BS; k0 += 4) {
      int ka = k0 + 2 * half;
      v2f ar, ai;
      ar.x = sO1[wid][l15][ka][0];
      ar.y = sO1[wid][l15][ka + 1][0];
      ai.x = sO1[wid][l15][ka][1];
      ai.y = sO1[wid][l15][ka + 1][1];
#pragma unroll
      for (int n = 0; n < 6; ++n) {
        int oc = n * 16 + l15;
        v2f wr, wi;
        wr.x = w2b[((size_t)ka * 96 + oc) * 2 + 0];
        wr.y = w2b[((size_t)(ka + 1) * 96 + oc) * 2 + 0];
        wi.x = w2b[((size_t)ka * 96 + oc) * 2 + 1];
        wi.y = w2b[((size_t)(ka + 1) * 96 + oc) * 2 + 1];
        v2f wn = -wi;
        dR[n] = wmma4(ar, wr, dR[n]);
        dR[n] = wmma4(ai, wn, dR[n]);
        dI[n] = wmma4(ai, wr, dI[n]);
        dI[n] = wmma4(ar, wi, dI[n]);
      }
    }
#pragma unroll
    for (int n = 0; n < 6; ++n) {
      int oc = n * 16 + l15;
      float br = sB2[oc * 2 + 0], bi = sB2[oc * 2 + 1];
      size_t co = (size_t)(cb + oc) * cfg::SH;
#pragma unroll
      for (int j = 0; j < 8; ++j) {
        int sIdx = s0 + j + 8 * half;
        if (sIdx < cfg::SH) {
          Zr[co + sIdx] = sshrink(dR[n][j] + br);
          Zi[co + sIdx] = sshrink(dI[n][j] + bi);
        }
      }
    }
    __syncthreads();
  }
}

// ---------------- stage 5: irFFT along W + residual add -----------------------
// in : planes (c, wf<91, h)  =>  A[(c*180+h)][wf] = Zr[c*SH + wf*180 + h]
// out: d_out[(c*180+h)*360 + w] = sum + x[...]
__global__ __launch_bounds__(256) void ifft_w_kernel(
    const float* __restrict__ Zr, const float* __restrict__ Zi,
    const float* __restrict__ bc, const float* __restrict__ bs,
    const float* __restrict__ x, float* __restrict__ out) {
  int gw = (int)((blockIdx.x * blockDim.x + threadIdx.x) >> 5);
  int mt = gw >> 2, g = gw & 3;  // 4 waves cover the 23 N tiles
  int lane = threadIdx.x & 31;
  int half = lane >> 4, l15 = lane & 15;
  int m0 = mt * 16;
  int m = m0 + l15;
  int c = m / cfg::H, h = m - c * cfg::H;
  const float* zr = Zr + (size_t)c * cfg::SH + h;
  const float* zi = Zi + (size_t)c * cfg::SH + h;
  v8f zero = {0.f, 0.f, 0.f, 0.f, 0.f, 0.f, 0.f, 0.f};
  v8f acc[6];
#pragma unroll
  for (int n = 0; n < 6; ++n) acc[n] = zero;
  for (int k0 = 0; k0 < 92; k0 += 4) {
    int ka = k0 + 2 * half;
    bool kv0 = ka < cfg::WF, kv1 = (ka + 1) < cfg::WF;
    v2f ar, ai;
    ar.x = kv0 ? zr[(size_t)ka * cfg::H] : 0.f;
    ar.y = kv1 ? zr[(size_t)(ka + 1) * cfg::H] : 0.f;
    ai.x = kv0 ? zi[(size_t)ka * cfg::H] : 0.f;
    ai.y = kv1 ? zi[(size_t)(ka + 1) * cfg::H] : 0.f;
#pragma unroll
    for (int n = 0; n < 6; ++n) {
      int nt = g * 6 + n;
      if (nt < 23) {  // wave-uniform guard
        int nc = nt * 16 + l15;
        v2f cc, ss;
        cc.x = bc[(size_t)ka * 368 + nc];
        cc.y = bc[(size_t)(ka + 1) * 368 + nc];
        ss.x = bs[(size_t)ka * 368 + nc];
        ss.y = bs[(size_t)(ka + 1) * 368 + nc];
        acc[n] = wmma4(ar, cc, acc[n]);
        acc[n] = wmma4(ai, ss, acc[n]);
      }
    }
  }
#pragma unroll
  for (int n = 0; n < 6; ++n) {
    int nt = g * 6 + n;
    if (nt < 23) {
      int w = nt * 16 + l15;
      if (w < cfg::W) {
#pragma unroll
        for (int j = 0; j < 8; ++j) {
          size_t o = (size_t)(m0 + j + 8 * half) * cfg::W + w;
          out[o] = acc[n][j] + x[o];
        }
      }
    }
  }
}

// ------------------------------------------------------------------------------
extern "C" void kernel_launch(void* const* d_in, const int* in_sizes, int n_in,
                              void* d_out, int out_size, void* d_ws,
                              size_t ws_size, hipStream_t stream) {
  (void)in_sizes; (void)n_in; (void)out_size;
  const float* x  = (const float*)d_in[0];
  const float* w1 = (const float*)d_in[1];
  const float* b1 = (const float*)d_in[2];
  const float* w2 = (const float*)d_in[3];
  const float* b2 = (const float*)d_in[4];
  float* out = (float*)d_out;
  float* ws = (float*)d_ws;
  if (ws_size < (size_t)cfg::TOTWS * sizeof(float)) return;

  float* B1c = ws + cfg::OB1C;  float* B1s = ws + cfg::OB1S;
  float* B2c = ws + cfg::OB2C;  float* B2s = ws + cfg::OB2S;
  float* B3c = ws + cfg::OB3C;  float* B3s = ws + cfg::OB3S;
  float* B4c = ws + cfg::OB4C;  float* B4s = ws + cfg::OB4S;
  float* P0r = ws + cfg::OP0R;  float* P0i = ws + cfg::OP0I;
  float* P1r = ws + cfg::OP1R;  float* P1i = ws + cfg::OP1I;

  init_basis_kernel<<<(34560 + 255) / 256, 256, 0, stream>>>(
      B1c, B1s, B2c, B2s, B3c, B3s, B4c, B4s);

  // rFFT along W:  138240/16 = 8640 waves, 8 waves/block
  fft_w_kernel<<<cfg::CH / 16 / 8, 256, 0, stream>>>(x, B1c, B1s, P0r, P0i);

  // forward DFT along H: 4368 M tiles * 2 N-halves = 8736 waves
  dft_h_kernel<0><<<(cfg::ROWH / 16) * 2 / 8, 256, 0, stream>>>(
      P0r, P0i, B2c, B2s, P1r, P1i);

  // block-diagonal complex MLP: 8 blocks * 128 tile groups
  mlp_kernel<<<cfg::NB * 128, 128, 0, stream>>>(P1r, P1i, w1, b1, w2, b2,
                                                P0r, P0i);

  // inverse DFT along H
  dft_h_kernel<1><<<(cfg::ROWH / 16) * 2 / 8, 256, 0, stream>>>(
      P0r, P0i, B3c, B3s, P1r, P1i);

  // irFFT along W + residual: 8640 M tiles * 4 N-groups = 34560 waves
  ifft_w_kernel<<<(cfg::CH / 16) * 4 / 8, 256, 0, stream>>>(P1r, P1i, B4c, B4s,
                                                            x, out);
}